// LinearSystem_52956946759997
// MI455X (gfx1250) — compile-verified
//
#include <hip/hip_runtime.h>
#include <hip/hip_bf16.h>

// ---------------------------------------------------------------------------
// HiPPO-LegS bilinear SSM scan on gfx1250 using V_WMMA_F32_16X16X4_F32.
// L=4096, BATCH*H=4096 rows, N=64, M=1.
// Kernel 1 (ssm_prep): per time-step, build fwd=(I+dt/2 A), invert (I-dt/2 A)
//   by Gauss-Jordan in LDS, form W = Mbar^T and Bbar, write W in a
//   WMMA-B-fragment-friendly layout.
// Kernel 2 (ssm_scan): 256 single-wave workgroups, each owning 16 state rows.
//   Per step: x_new(16x64) = x(16x64) @ W(64x64) + u*Bbar via 68 f32 WMMAs,
//   state round-trips through a 4KB LDS tile (D-layout -> A-layout).
// ---------------------------------------------------------------------------

typedef __attribute__((ext_vector_type(2))) float v2f;
typedef __attribute__((ext_vector_type(8))) float v8f;

__device__ __forceinline__ v8f wmma_f32_k4(v2f a, v2f b, v8f c) {
  // 8 args: (neg_a, A, neg_b, B, c_mod, C, reuse_a, reuse_b)
  return __builtin_amdgcn_wmma_f32_16x16x4_f32(
      false, a, false, b, (short)0, c, false, false);
}

#define ROWS_TOTAL 4096   // BATCH*H
#define NSTATE 64
#define LSEQ 4096

// ---------------------------------------------------------------------------
// Precompute kernel: one 64-thread workgroup per time step in the chunk.
// frag layout per step (4096 floats):  frag[((kk*4 + j)*32 + lane)*2 + comp]
//   = W[4*kk + 2*(lane>>4) + comp][16*j + (lane&15)],  W = Mbar^T.
// ---------------------------------------------------------------------------
__global__ void ssm_prep(const float* __restrict__ dt,
                         const float* __restrict__ A,
                         const float* __restrict__ Bvec,
                         float* __restrict__ frag,
                         float* __restrict__ Bb,
                         int t0) {
  __shared__ float aug[64 * 128];   // [ (I - dt/2 A) | I ]  -> [ I | inv ]
  __shared__ float fwd[64 * 64];    //   I + dt/2 A
  const int n  = threadIdx.x;       // 0..63, owns row n
  const int tt = blockIdx.x;
  const int t  = t0 + tt;
  const float hdt = 0.5f * dt[t];

  for (int c = 0; c < 64; ++c) {
    const float a   = A[n * 64 + c];
    const float eye = (n == c) ? 1.0f : 0.0f;
    fwd[n * 64 + c]       = eye + hdt * a;
    aug[n * 128 + c]      = eye - hdt * a;
    aug[n * 128 + 64 + c] = eye;
  }
  __syncthreads();

  // Gauss-Jordan, no pivoting (diagonally dominant: |hdt*A| << 1 off-diag).
  for (int p = 0; p < 64; ++p) {
    if (n == p) {
      const float pinv = 1.0f / aug[p * 128 + p];
      for (int c = 0; c < 128; ++c) aug[p * 128 + c] *= pinv;
    }
    __syncthreads();
    if (n != p) {
      const float f = aug[n * 128 + p];
      for (int c = 0; c < 128; ++c) aug[n * 128 + c] -= f * aug[p * 128 + c];
    }
    __syncthreads();
  }

  // Mbar[n][k] = sum_j inv[n][j] * fwd[j][k];  W[k][n] = Mbar[n][k].
  float* fragT = frag + (size_t)tt * 4096;
  const int j = n >> 4, c = n & 15;
  for (int k = 0; k < 64; ++k) {
    float s = 0.0f;
    for (int q = 0; q < 64; ++q)
      s += aug[n * 128 + 64 + q] * fwd[q * 64 + k];
    const int kk = k >> 2, h = (k >> 1) & 1, comp = k & 1;
    const int lane = h * 16 + c;
    fragT[((kk * 4 + j) * 32 + lane) * 2 + comp] = s;
  }

  // Bbar[n] = dt * sum_j inv[n][j] * B[j]
  float sb = 0.0f;
  for (int q = 0; q < 64; ++q)
    sb += aug[n * 128 + 64 + q] * Bvec[q];
  Bb[(size_t)tt * 64 + n] = 2.0f * hdt * sb;
}

// ---------------------------------------------------------------------------
// Scan kernel: 256 workgroups x 32 threads (1 wave). Wave owns rows
// [16*blockIdx.x, 16*blockIdx.x+16). State tile 16x64 f32 in LDS.
// ---------------------------------------------------------------------------
__global__ void ssm_scan(const float* __restrict__ frag,
                         const float* __restrict__ Bb,
                         float* __restrict__ Xst,
                         const float* __restrict__ u,
                         const float* __restrict__ C,
                         const float* __restrict__ Dv,
                         float* __restrict__ y,
                         float* __restrict__ xfin,
                         int t0, int Tc, int first, int last) {
  __shared__ float xs[16 * 64];
  __shared__ float cs[64];
  const int l    = threadIdx.x;        // lane 0..31
  const int row0 = blockIdx.x * 16;
  const int r = l & 15, h = l >> 4;

  for (int i = l; i < 16 * 64; i += 32)
    xs[i] = first ? 0.0f : Xst[(size_t)row0 * 64 + i];
  cs[l]      = C[l];
  cs[l + 32] = C[l + 32];
  const float D0 = Dv[0];
  __syncthreads();

  const v8f vzero = {0.f, 0.f, 0.f, 0.f, 0.f, 0.f, 0.f, 0.f};

  for (int tt = 0; tt < Tc; ++tt) {
    const int t = t0 + tt;
    const float u_t = (l < 16) ? u[(size_t)t * ROWS_TOTAL + row0 + l] : 0.0f;

    v8f acc[4];
#pragma unroll
    for (int j = 0; j < 4; ++j) acc[j] = vzero;

    const float* fb = frag + (size_t)tt * 4096;
#pragma unroll
    for (int kk = 0; kk < 16; ++kk) {
      // A fragment from LDS state: lane holds row r, K = 4*kk + 2*h + {0,1}
      const v2f a = *(const v2f*)&xs[r * 64 + kk * 4 + 2 * h];
#pragma unroll
      for (int j = 0; j < 4; ++j) {
        const v2f b = *(const v2f*)&fb[((kk * 4 + j) * 32 + l) * 2];
        acc[j] = wmma_f32_k4(a, b, acc[j]);
      }
    }

    // rank-1 input term: x += u * Bbar  (K=0 column of A, K=0 row of B)
    v2f au; au.x = u_t; au.y = 0.0f;
#pragma unroll
    for (int j = 0; j < 4; ++j) {
      v2f bu;
      bu.x = (l < 16) ? Bb[(size_t)tt * 64 + 16 * j + r] : 0.0f;
      bu.y = 0.0f;
      acc[j] = wmma_f32_k4(au, bu, acc[j]);
    }

    __syncthreads();
    // D layout -> LDS row-major: VGPR v, lane l = elem (row v+8h, col 16j+r)
#pragma unroll
    for (int j = 0; j < 4; ++j)
#pragma unroll
      for (int v = 0; v < 8; ++v)
        xs[(v + 8 * h) * 64 + 16 * j + r] = acc[j][v];
    __syncthreads();

    // y[t,row] = x_new[row,:] . C[0,:] + u*D0   (lane pair (r,h) splits cols)
    float part = 0.0f;
#pragma unroll
    for (int q = 0; q < 8; ++q) {
      const float4 xv = *(const float4*)&xs[r * 64 + h * 32 + q * 4];
      const float4 cv = *(const float4*)&cs[h * 32 + q * 4];
      part += xv.x * cv.x + xv.y * cv.y + xv.z * cv.z + xv.w * cv.w;
    }
    part += __shfl_xor(part, 16, 32);
    if (l < 16)
      y[(size_t)t * ROWS_TOTAL + row0 + l] = part + u_t * D0;
  }

  __syncthreads();
  for (int i = l; i < 16 * 64; i += 32) {
    Xst[(size_t)row0 * 64 + i] = xs[i];
    if (last) xfin[(size_t)row0 * 64 + i] = xs[i];
  }
}

// ---------------------------------------------------------------------------
extern "C" void kernel_launch(void* const* d_in, const int* in_sizes, int n_in,
                              void* d_out, int out_size, void* d_ws, size_t ws_size,
                              hipStream_t stream) {
  const float* dt = (const float*)d_in[0];  // (L,)
  const float* u  = (const float*)d_in[1];  // (L, BATCH, H)
  const float* A  = (const float*)d_in[2];  // (N, N)
  const float* B  = (const float*)d_in[3];  // (N,)
  const float* C  = (const float*)d_in[4];  // (M, N) = (1, 64)
  const float* Dv = (const float*)d_in[5];  // (M,) = (1,)

  float* y    = (float*)d_out;                              // L*4096 floats
  float* xfin = y + (size_t)LSEQ * ROWS_TOTAL;              // 4096*64 floats

  // Workspace: [ Xstate (4096*64) | Bbar (Tc*64) | frags (Tc*4096) ]
  float* ws  = (float*)d_ws;
  float* Xst = ws;
  const long long xst_f   = (long long)ROWS_TOTAL * NSTATE;           // 262144
  const long long avail_f = (long long)(ws_size / 4) - xst_f;
  long long Tc_ll = avail_f / (4096 + 64);
  if (Tc_ll > 512) Tc_ll = 512;
  if (Tc_ll < 1)   Tc_ll = 1;
  const int Tc = (int)Tc_ll;

  float* Bb   = ws + xst_f;
  float* frag = Bb + (size_t)Tc * 64;

  const int nch = (LSEQ + Tc - 1) / Tc;
  for (int ch = 0; ch < nch; ++ch) {
    const int t0  = ch * Tc;
    const int cur = (LSEQ - t0 < Tc) ? (LSEQ - t0) : Tc;
    ssm_prep<<<cur, 64, 0, stream>>>(dt, A, B, frag, Bb, t0);
    ssm_scan<<<256, 32, 0, stream>>>(frag, Bb, Xst, u, C, Dv, y, xfin,
                                     t0, cur, ch == 0 ? 1 : 0,
                                     ch == nch - 1 ? 1 : 0);
  }
}